// Genetation_42210938585609
// MI455X (gfx1250) — compile-verified
//
#include <hip/hip_runtime.h>
#include <hip/hip_bf16.h>

typedef float v2f __attribute__((ext_vector_type(2)));
typedef float v8f __attribute__((ext_vector_type(8)));

#define FDIM 128
#define NEG_SLOPE 0.2f

// ---------- helpers ----------
__device__ __forceinline__ unsigned int f2mono(float f) {
    unsigned int b = __float_as_uint(f);
    // monotone map: float order -> unsigned order
    return b ^ ((unsigned int)((int)b >> 31) | 0x80000000u);
}
__device__ __forceinline__ float mono2f(unsigned int u) {
    unsigned int b = (u & 0x80000000u) ? (u ^ 0x80000000u) : ~u;
    return __uint_as_float(b);
}

// ---------- init: zero out accumulator, emax, denom ----------
__global__ void init_kernel(float* __restrict__ out, unsigned int* __restrict__ emax_u,
                            float* __restrict__ denom, int N) {
    int i = blockIdx.x * blockDim.x + threadIdx.x;
    if (i < N * FDIM) out[i] = 0.0f;
    if (i < N) { emax_u[i] = 0u; denom[i] = 0.0f; }
}

// ---------- GEMM h = x @ W via fp32 WMMA 16x16x4 ----------
// grid.x = ceil(N/16); blockDim = 256 (8 waves); wave w owns N-tile [w*16, w*16+16)
__global__ void gemm_h_kernel(const float* __restrict__ x, const float* __restrict__ W,
                              float* __restrict__ h, int N) {
    const int lane = threadIdx.x & 31;
    const int wave = threadIdx.x >> 5;           // 0..7 -> column tile
    const int m0   = blockIdx.x * 16;
    const int n0   = wave * 16;
    const int row  = lane & 15;                   // A: M row for this lane
    const int col  = lane & 15;                   // B/C/D: N col for this lane
    const int kOff = (lane >> 4) << 1;            // lanes 16-31 hold K+2,K+3
    const bool fullTile = (m0 + 16 <= N);         // uniform across the block

    int mrow = m0 + row;
    if (!fullTile && mrow >= N) mrow = N - 1;     // clamp only in tail tile
    const float* __restrict__ xrow = x + (size_t)mrow * FDIM;

    v8f c = {};
#pragma unroll 8
    for (int kb = 0; kb < FDIM / 4; ++kb) {
        const int k = kb * 4 + kOff;
        v2f a, b;
        a.x = xrow[k];
        a.y = xrow[k + 1];
        b.x = W[(size_t)k * FDIM + n0 + col];
        b.y = W[(size_t)(k + 1) * FDIM + n0 + col];
        // D = A(16x4,f32) * B(4x16,f32) + C ; exact fp32 on the matrix pipe
        c = __builtin_amdgcn_wmma_f32_16x16x4_f32(false, a, false, b, (short)0, c,
                                                  false, false);
    }

    const int rbase = (lane >> 4) << 3;           // lanes 16-31 hold M rows 8..15
    float* __restrict__ hp = h + (size_t)(m0 + rbase) * FDIM + n0 + col;
    if (fullTile) {
        // straight-line stores: uniform branch, no per-lane EXEC juggling
#pragma unroll
        for (int r = 0; r < 8; ++r) {
            hp[(size_t)r * FDIM] = c[r];
        }
    } else {
#pragma unroll
        for (int r = 0; r < 8; ++r) {
            const int m = m0 + rbase + r;
            if (m < N) h[(size_t)m * FDIM + n0 + col] = c[r];
        }
    }
}

// ---------- per-node attention dots: a_src[n]=h[n].att_src, a_dst[n]=h[n].att_dst ----------
// one wave per node; lane covers 4 features
__global__ void attn_dots_kernel(const float* __restrict__ h,
                                 const float* __restrict__ att_src,
                                 const float* __restrict__ att_dst,
                                 float* __restrict__ a_src, float* __restrict__ a_dst, int N) {
    const int wavesPerBlock = blockDim.x >> 5;
    const int node = blockIdx.x * wavesPerBlock + (threadIdx.x >> 5);
    if (node >= N) return;
    const int lane = threadIdx.x & 31;
    const float4 hv = ((const float4*)(h + (size_t)node * FDIM))[lane];
    const float4 as = ((const float4*)att_src)[lane];
    const float4 ad = ((const float4*)att_dst)[lane];
    float s = hv.x * as.x + hv.y * as.y + hv.z * as.z + hv.w * as.w;
    float d = hv.x * ad.x + hv.y * ad.y + hv.z * ad.z + hv.w * ad.w;
#pragma unroll
    for (int off = 16; off > 0; off >>= 1) {
        s += __shfl_xor(s, off, 32);
        d += __shfl_xor(d, off, 32);
    }
    if (lane == 0) { a_src[node] = s; a_dst[node] = d; }
}

// ---------- edge pass 1: e = leaky_relu(a_src[src]+a_dst[dst]); segment max ----------
__global__ void edge_max_kernel(const long long* __restrict__ srcI,
                                const long long* __restrict__ dstI,
                                const float* __restrict__ a_src,
                                const float* __restrict__ a_dst,
                                unsigned int* __restrict__ emax_u,
                                float* __restrict__ e_buf, int E) {
    int i = blockIdx.x * blockDim.x + threadIdx.x;
    if (i >= E) return;
    const int s = (int)srcI[i];
    const int d = (int)dstI[i];
    float e = a_src[s] + a_dst[d];
    e = (e > 0.0f) ? e : NEG_SLOPE * e;
    e_buf[i] = e;
    atomicMax(emax_u + d, f2mono(e));            // exact f32 max via monotone u32
}

// ---------- edge pass 2: e_exp = exp(e - max[dst]); segment sum ----------
__global__ void edge_exp_kernel(const long long* __restrict__ dstI,
                                const unsigned int* __restrict__ emax_u,
                                float* __restrict__ e_buf,
                                float* __restrict__ denom, int E) {
    int i = blockIdx.x * blockDim.x + threadIdx.x;
    if (i >= E) return;
    const int d = (int)dstI[i];
    const float em = mono2f(emax_u[d]);          // finite whenever node has edges
    const float ex = __expf(e_buf[i] - em);
    e_buf[i] = ex;
    atomicAdd(denom + d, ex);                    // global_atomic_add_f32
}

// ---------- edge pass 3: out[dst] += alpha * h[src]  (wave per edge) ----------
__global__ void scatter_kernel(const long long* __restrict__ srcI,
                               const long long* __restrict__ dstI,
                               const float* __restrict__ e_exp,
                               const float* __restrict__ denom,
                               const float* __restrict__ h,
                               float* __restrict__ out, int E) {
    const int wavesPerBlock = blockDim.x >> 5;
    const long long eid = (long long)blockIdx.x * wavesPerBlock + (threadIdx.x >> 5);
    if (eid >= E) return;
    const int lane = threadIdx.x & 31;
    const int s = (int)srcI[eid];
    const int d = (int)dstI[eid];
    const float alpha = e_exp[eid] / (denom[d] + 1e-16f);
    const float4 hv = ((const float4*)(h + (size_t)s * FDIM))[lane];
    float* o = out + (size_t)d * FDIM + lane * 4;
    atomicAdd(o + 0, alpha * hv.x);
    atomicAdd(o + 1, alpha * hv.y);
    atomicAdd(o + 2, alpha * hv.z);
    atomicAdd(o + 3, alpha * hv.w);
}

// ---------- final: out = relu(out + bias) ----------
__global__ void bias_relu_kernel(float* __restrict__ out, const float* __restrict__ bias,
                                 int total) {
    int i = blockIdx.x * blockDim.x + threadIdx.x;
    if (i >= total) return;
    float v = out[i] + bias[i & (FDIM - 1)];
    out[i] = v > 0.0f ? v : 0.0f;
}

extern "C" void kernel_launch(void* const* d_in, const int* in_sizes, int n_in,
                              void* d_out, int out_size, void* d_ws, size_t ws_size,
                              hipStream_t stream) {
    const float*     x       = (const float*)d_in[0];
    const long long* ei      = (const long long*)d_in[1];   // int64 [2, E]
    const float*     W       = (const float*)d_in[2];
    const float*     att_src = (const float*)d_in[3];
    const float*     att_dst = (const float*)d_in[4];
    const float*     bias    = (const float*)d_in[5];

    const int N = in_sizes[0] / FDIM;
    const int E = in_sizes[1] / 2;
    const long long* srcI = ei;
    const long long* dstI = ei + E;

    // workspace layout (16B aligned slices)
    char* ws = (char*)d_ws;
    float*        h      = (float*)ws;                 ws += (size_t)N * FDIM * sizeof(float);
    float*        a_src  = (float*)ws;                 ws += (size_t)N * sizeof(float);
    float*        a_dst  = (float*)ws;                 ws += (size_t)N * sizeof(float);
    unsigned int* emax_u = (unsigned int*)ws;          ws += (size_t)N * sizeof(unsigned int);
    float*        denom  = (float*)ws;                 ws += (size_t)N * sizeof(float);
    float*        e_buf  = (float*)ws;                 // E floats

    float* out = (float*)d_out;

    const int BT = 256;

    // 1) init accumulators
    init_kernel<<<(N * FDIM + BT - 1) / BT, BT, 0, stream>>>(out, emax_u, denom, N);

    // 2) h = x @ W  (fp32 WMMA, one 16x16 tile per wave, 8 waves = Fout 128)
    gemm_h_kernel<<<(N + 15) / 16, BT, 0, stream>>>(x, W, h, N);

    // 3) per-node attention projections (wave per node)
    attn_dots_kernel<<<(N + 7) / 8, BT, 0, stream>>>(h, att_src, att_dst, a_src, a_dst, N);

    // 4) edge passes: max -> exp/sum -> weighted scatter
    edge_max_kernel<<<(E + BT - 1) / BT, BT, 0, stream>>>(srcI, dstI, a_src, a_dst,
                                                          emax_u, e_buf, E);
    edge_exp_kernel<<<(E + BT - 1) / BT, BT, 0, stream>>>(dstI, emax_u, e_buf, denom, E);
    scatter_kernel<<<(E + 7) / 8, BT, 0, stream>>>(srcI, dstI, e_buf, denom, h, out, E);

    // 5) bias + relu
    bias_relu_kernel<<<(N * FDIM + BT - 1) / BT, BT, 0, stream>>>(out, bias, N * FDIM);
}